// SAGE_39170101740217
// MI455X (gfx1250) — compile-verified
//
#include <hip/hip_runtime.h>

// ---------------------------------------------------------------------------
// GraphSAGE 2-layer forward on MI455X (gfx1250, wave32).
//   layer: out = (mean_aggr) @ Wl + x @ Wr + b   (relu after layer 1)
// Irregular aggregation via native fp32 global atomics (L2-resident tables),
// dense 64-wide projections via V_WMMA_F32_16X16X4_F32.
// ---------------------------------------------------------------------------

typedef __attribute__((ext_vector_type(2))) float v2f;
typedef __attribute__((ext_vector_type(8))) float v8f;

#define GN_NODES 50000
#define GN_EDGES 800000
#define GF1 96
#define GF2 64

// ---- zero-fill scratch (cnt | aggr1 | aggr2 are contiguous) ---------------
__global__ void sage_zero_kernel(float* __restrict__ p, int n) {
    int i = blockIdx.x * blockDim.x + threadIdx.x;
    if (i < n) p[i] = 0.0f;
}

// ---- edge scatter: aggr[tgt] += x[src]; optional degree count --------------
// One thread per (edge, feature). Consecutive lanes cover consecutive
// features of one source row -> coalesced gather reads; atomic adds land on
// consecutive addresses of the target row and resolve in L2.
__global__ void sage_scatter_kernel(const float* __restrict__ x,
                                    const int* __restrict__ src,
                                    const int* __restrict__ tgt,
                                    float* __restrict__ aggr,
                                    float* __restrict__ cnt,   // nullptr => skip
                                    int E, int F) {
    int i = blockIdx.x * blockDim.x + threadIdx.x;
    if (i >= E * F) return;
    int e = i / F;
    int f = i - e * F;
    int s = src[e];
    int t = tgt[e];
    unsafeAtomicAdd(&aggr[(size_t)t * F + f], x[(size_t)s * F + f]);
    if (cnt != nullptr && f == 0) {
        unsafeAtomicAdd(&cnt[t], 1.0f);
    }
}

// ---- dense layer: out = (aggr/deg) @ Wl + x @ Wr + bias, optional relu -----
// Block = 128 threads = 4 wave32s. Block owns 16 node-rows; wave w owns
// output columns [16w, 16w+16). K stepped by 4 with V_WMMA_F32_16X16X4_F32.
//
// VGPR layouts per CDNA5 ISA 7.12.2 (wave32):
//   A 16x4 : lane L holds row M=L%16; VGPR0/1 = K={0,1} (lanes 0-15),
//            K={2,3} (lanes 16-31)
//   B 4x16 : lane L holds col N=L%16; VGPR0/1 = K={0,1}/{2,3} by lane half
//   C/D    : VGPR v holds M=v (lanes 0-15), M=v+8 (lanes 16-31), N=L%16
__global__ void __launch_bounds__(128)
sage_layer_kernel(const float* __restrict__ aggr,
                  const float* __restrict__ cnt,
                  const float* __restrict__ x,
                  const float* __restrict__ Wl,
                  const float* __restrict__ Wr,
                  const float* __restrict__ bias,
                  float* __restrict__ out,
                  int K, int do_relu) {
    const int lane = threadIdx.x & 31;
    const int wave = threadIdx.x >> 5;
    const int half = lane >> 4;       // 0: K lo pair, 1: K hi pair
    const int l16  = lane & 15;
    const int rowA = blockIdx.x * 16 + l16;   // A-matrix row this lane feeds
    const int col  = wave * 16 + l16;         // B / D column this lane owns

    const float invdeg = 1.0f / fmaxf(cnt[rowA], 1.0f);

    const float* __restrict__ arow = aggr + (size_t)rowA * K;
    const float* __restrict__ xrow = x    + (size_t)rowA * K;

    v8f c = {0.f, 0.f, 0.f, 0.f, 0.f, 0.f, 0.f, 0.f};

    // term 1: normalized neighbor mean through Wl
    for (int k = 0; k < K; k += 4) {
        const int ka = k + 2 * half;          // even -> 8B-aligned pair load
        v2f a = *(const v2f*)(arow + ka);
        a.x *= invdeg;
        a.y *= invdeg;
        v2f b;
        b.x = Wl[(size_t)ka * 64 + col];
        b.y = Wl[(size_t)(ka + 1) * 64 + col];
        c = __builtin_amdgcn_wmma_f32_16x16x4_f32(false, a, false, b,
                                                  (short)0, c, false, false);
    }
    // term 2: self features through Wr
    for (int k = 0; k < K; k += 4) {
        const int ka = k + 2 * half;
        v2f a = *(const v2f*)(xrow + ka);
        v2f b;
        b.x = Wr[(size_t)ka * 64 + col];
        b.y = Wr[(size_t)(ka + 1) * 64 + col];
        c = __builtin_amdgcn_wmma_f32_16x16x4_f32(false, a, false, b,
                                                  (short)0, c, false, false);
    }

    const float bv = bias[col];
#pragma unroll
    for (int v = 0; v < 8; ++v) {
        const int row = blockIdx.x * 16 + v + 8 * half;
        float val = c[v] + bv;
        if (do_relu) val = fmaxf(val, 0.0f);
        out[(size_t)row * 64 + col] = val;
    }
}

// ---------------------------------------------------------------------------
extern "C" void kernel_launch(void* const* d_in, const int* in_sizes, int n_in,
                              void* d_out, int out_size, void* d_ws, size_t ws_size,
                              hipStream_t stream) {
    const float* x   = (const float*)d_in[0];
    const int*   ei  = (const int*)  d_in[1];
    const float* Wl1 = (const float*)d_in[2];
    const float* Wr1 = (const float*)d_in[3];
    const float* b1  = (const float*)d_in[4];
    const float* Wl2 = (const float*)d_in[5];
    const float* Wr2 = (const float*)d_in[6];
    const float* b2  = (const float*)d_in[7];
    float* out = (float*)d_out;

    const int* src = ei;             // edge_index[0, :]
    const int* tgt = ei + GN_EDGES;  // edge_index[1, :]

    // workspace layout (floats): cnt | aggr1 | aggr2 | h   (~45 MB total)
    float* ws    = (float*)d_ws;
    float* cnt   = ws;                                  // GN_NODES
    float* aggr1 = cnt + GN_NODES;                      // GN_NODES * 96
    float* aggr2 = aggr1 + (size_t)GN_NODES * GF1;      // GN_NODES * 64
    float* h     = aggr2 + (size_t)GN_NODES * GF2;      // GN_NODES * 64

    // 1) zero cnt + aggr1 + aggr2 (contiguous region)
    const int zn = GN_NODES * (1 + GF1 + GF2);
    sage_zero_kernel<<<(zn + 255) / 256, 256, 0, stream>>>(cnt, zn);

    // 2) layer-1 neighbor sum + degree
    {
        const int total = GN_EDGES * GF1;
        sage_scatter_kernel<<<(total + 255) / 256, 256, 0, stream>>>(
            x, src, tgt, aggr1, cnt, GN_EDGES, GF1);
    }

    // 3) h = relu(aggr1/deg @ Wl1 + x @ Wr1 + b1)
    sage_layer_kernel<<<GN_NODES / 16, 128, 0, stream>>>(
        aggr1, cnt, x, Wl1, Wr1, b1, h, GF1, 1);

    // 4) layer-2 neighbor sum (degrees unchanged)
    {
        const int total = GN_EDGES * GF2;
        sage_scatter_kernel<<<(total + 255) / 256, 256, 0, stream>>>(
            h, src, tgt, aggr2, nullptr, GN_EDGES, GF2);
    }

    // 5) out = aggr2/deg @ Wl2 + h @ Wr2 + b2
    sage_layer_kernel<<<GN_NODES / 16, 128, 0, stream>>>(
        aggr2, cnt, h, Wl2, Wr2, b2, out, GF2, 0);
}